// LightGCN_layer_23493471109149
// MI455X (gfx1250) — compile-verified
//
#include <hip/hip_runtime.h>

typedef float v2f __attribute__((ext_vector_type(2)));
typedef float v8f __attribute__((ext_vector_type(8)));

#define DIM 64
#define WSTRIDE 80   // LDS row stride in floats: 2*80 mod 64 = 32 -> lane halves hit disjoint banks

// ---------------------------------------------------------------------------
// Kernel 1: zero the accumulator (d_out doubles as agg buffer; 128 MB < L2)
// ---------------------------------------------------------------------------
__global__ void zero_kernel(float4* __restrict__ p, long long n4) {
    long long i = (long long)blockIdx.x * blockDim.x + threadIdx.x;
    long long stride = (long long)gridDim.x * blockDim.x;
    float4 z = make_float4(0.f, 0.f, 0.f, 0.f);
    for (; i < n4; i += stride) p[i] = z;
}

// ---------------------------------------------------------------------------
// Kernel 2: edge scatter. One wave32 per edge: coalesced 256B source-row read
// (float2 per lane), scaled, accumulated with f32 global atomics that land in
// the 192MB L2 (agg is 128MB, fully resident).
// ---------------------------------------------------------------------------
__global__ __launch_bounds__(256) void scatter_kernel(
    const float* __restrict__ users,
    const float* __restrict__ items,
    const float* __restrict__ vals,
    const int*   __restrict__ rows,
    const int*   __restrict__ cols,
    float*       __restrict__ agg,
    int nnz, int num_user) {
    int lane = threadIdx.x & 31;
    long long wave = ((long long)blockIdx.x * blockDim.x + threadIdx.x) >> 5;
    if (wave >= nnz) return;
    int e = (int)wave;
    float v = vals[e];
    int r = rows[e];
    int c = cols[e];
    const float* src = (c < num_user) ? (users + (long long)c * DIM)
                                      : (items + (long long)(c - num_user) * DIM);
    float2 x = ((const float2*)src)[lane];
    float* dst = agg + (long long)r * DIM + 2 * lane;
    unsafeAtomicAdd(dst,     x.x * v);   // global_atomic_add_f32
    unsafeAtomicAdd(dst + 1, x.y * v);
}

// ---------------------------------------------------------------------------
// Kernel 3: in-place [rows x 64] @ [64 x 64] + leaky_relu(0.2) via
// V_WMMA_F32_16X16X4_F32. One wave owns a 16x64 tile (4 accumulators of
// 16x16), K=64 in 16 steps of K=4. Safe in place: each output row depends
// only on the same agg row, and a wave reads all its rows before writing.
//
// Fragment layouts (ISA 7.12.2, 32-bit):
//   A 16x4 : lanes 0-15 -> K=k,k+1 ; lanes 16-31 -> K=k+2,k+3 ; M = lane%16
//   B 4x16 : VGPR0/1 = K rows (k+2*hi, k+2*hi+1), N = lane%16 per half
//   C 16x16: vgpr r -> M = r + 8*(lane/16), N = lane%16
// ---------------------------------------------------------------------------
__global__ __launch_bounds__(256) void gemm_leaky_wmma(
    float* __restrict__ agg,            // in: agg, out: leaky_relu(agg @ w)
    const float* __restrict__ w,        // 64 x 64 row-major
    int nrows) {
    __shared__ float wlds[DIM * WSTRIDE];

    for (int i = threadIdx.x; i < DIM * DIM; i += blockDim.x) {
        int r = i >> 6, c = i & 63;
        wlds[r * WSTRIDE + c] = w[i];
    }
    __syncthreads();

    int waveId = threadIdx.x >> 5;
    int lane   = threadIdx.x & 31;
    int lo     = lane & 15;
    int hi     = lane >> 4;

    long long rowBase = ((long long)blockIdx.x * 8 + waveId) * 16;
    if (rowBase >= nrows) return;   // nrows is a multiple of 16

    v8f acc[4];
    #pragma unroll
    for (int t = 0; t < 4; ++t) acc[t] = (v8f){0,0,0,0,0,0,0,0};

    const float* arow = agg + (rowBase + lo) * DIM;

    #pragma unroll
    for (int k = 0; k < DIM; k += 4) {
        int ka = k + 2 * hi;                 // this lane-half's K pair
        v2f a;
        a.x = arow[ka];
        a.y = arow[ka + 1];

        #pragma unroll
        for (int t = 0; t < 4; ++t) {
            int n = t * 16 + lo;
            v2f b;
            b.x = wlds[ka * WSTRIDE + n];
            b.y = wlds[(ka + 1) * WSTRIDE + n];
            acc[t] = __builtin_amdgcn_wmma_f32_16x16x4_f32(
                /*neg_a=*/false, a, /*neg_b=*/false, b,
                /*c_mod=*/(short)0, acc[t],
                /*reuse_a=*/false, /*reuse_b=*/false);
        }
    }

    // leaky_relu + store (all agg reads for these 16 rows are done)
    #pragma unroll
    for (int t = 0; t < 4; ++t) {
        #pragma unroll
        for (int r = 0; r < 8; ++r) {
            float x = acc[t][r];
            x = (x > 0.f) ? x : 0.2f * x;
            agg[(rowBase + r + 8 * hi) * DIM + t * 16 + lo] = x;
        }
    }
}

// ---------------------------------------------------------------------------
extern "C" void kernel_launch(void* const* d_in, const int* in_sizes, int n_in,
                              void* d_out, int out_size, void* d_ws, size_t ws_size,
                              hipStream_t stream) {
    const float* users = (const float*)d_in[0];
    const float* items = (const float*)d_in[1];
    const float* vals  = (const float*)d_in[2];
    const float* w     = (const float*)d_in[3];
    const int*   rows  = (const int*)d_in[4];
    const int*   cols  = (const int*)d_in[5];
    float* out = (float*)d_out;

    int num_user = in_sizes[0] / DIM;
    int num_item = in_sizes[1] / DIM;
    int n        = num_user + num_item;     // 500000, multiple of 16
    int nnz      = in_sizes[2];

    // 1) zero accumulator
    long long n4 = (long long)n * DIM / 4;
    zero_kernel<<<4096, 256, 0, stream>>>((float4*)out, n4);

    // 2) scatter: one wave per edge, 8 waves per 256-thread block
    long long waves = nnz;
    int sblocks = (int)((waves + 7) / 8);
    scatter_kernel<<<sblocks, 256, 0, stream>>>(users, items, vals, rows, cols,
                                                out, nnz, num_user);

    // 3) WMMA GEMM + leaky relu, in place; 8 waves x 16 rows = 128 rows/block
    int gblocks = (n / 16 + 7) / 8;
    gemm_leaky_wmma<<<gblocks, 256, 0, stream>>>(out, w, n);
}